// BaseOpenSetClassifier_24945170055185
// MI455X (gfx1250) — compile-verified
//
#include <hip/hip_runtime.h>

// Problem constants (match reference)
#define BDIM   16
#define NPIX   16384
#define DDIM   64
#define KTPL   64
#define NWAVES 4
#define THREADS (NWAVES * 32)
#define PIX_PER_WG 16
#define LDSW   68            // padded row stride (floats): bank-conflict-free WMMA reads
#define LDSW4  17            // in float4 units

typedef float v2f __attribute__((ext_vector_type(2)));
typedef float v8f __attribute__((ext_vector_type(8)));
typedef int   v4i __attribute__((ext_vector_type(4)));

// ---------------- async global->LDS staging (CDNA5 ASYNCcnt path) ------------
#if __has_builtin(__builtin_amdgcn_global_load_async_to_lds_b128) && \
    __has_builtin(__builtin_amdgcn_s_wait_asynccnt)
#define USE_ASYNC 1
#else
#define USE_ASYNC 0
#endif

#if USE_ASYNC
typedef __attribute__((address_space(1))) v4i gv4i;   // prints as "int4 __device__*"
typedef __attribute__((address_space(3))) v4i lv4i;   // prints as "int4 __shared__*"
__device__ __forceinline__ void cp16(const float* g, float* l) {
  // per-lane 16B DMA into LDS; LDS offset = low 32 bits of the flat address
  __builtin_amdgcn_global_load_async_to_lds_b128(
      (gv4i*)(uintptr_t)g,
      (lv4i*)(unsigned int)(uintptr_t)l,
      /*offset=*/0, /*cpol=*/0);
}
__device__ __forceinline__ void stage_wait() { __builtin_amdgcn_s_wait_asynccnt(0); }
#else
__device__ __forceinline__ void cp16(const float* g, float* l) {
  *(float4*)l = *(const float4*)g;
}
__device__ __forceinline__ void stage_wait() {}
#endif

// Stage NF4*THREADS float4 tiles: 16 float4 per 64-float row, padded LDS rows.
// Consecutive lanes cover one full 256B global row -> perfectly coalesced.
template <int NF4>
__device__ __forceinline__ void stage(const float* __restrict__ gbase,
                                      float* lbuf, int n, int tid) {
#pragma unroll
  for (int i = 0; i < NF4; ++i) {
    const int f   = tid + THREADS * i;
    const int row = f >> 4;
    const int c4  = f & 15;
    cp16(gbase + ((size_t)row * NPIX + n) * DDIM + c4 * 4,
         lbuf + row * LDSW + c4 * 4);
  }
}

__global__ __launch_bounds__(THREADS)
void openset_classify_kernel(const float* __restrict__ frame,  // [B, N, D]
                             const float* __restrict__ tmpl,   // [K, N, D]
                             const int*   __restrict__ tcls,   // [K]
                             float* __restrict__ out)          // m0|m1|min|pred [B,N] each
{
  __shared__ float x_lds[2][BDIM * LDSW];     // double-buffered X tiles
  __shared__ float t_lds[2][KTPL * LDSW];     // double-buffered T tiles
  __shared__ float sums[BDIM + KTPL];         // x2[0..15], t2[16..79]
  __shared__ float dist_lds[BDIM * LDSW];     // padded [b][k] partial distances

  const int tid  = threadIdx.x;
  const int wave = tid >> 5;   // template tile 0..3
  const int lane = tid & 31;
  const int j    = lane & 15;
  const int hi   = lane >> 4;

  float* mask0 = out;
  float* mask1 = out + (size_t)BDIM * NPIX;
  float* mind  = out + (size_t)2 * BDIM * NPIX;
  float* pred  = out + (size_t)3 * BDIM * NPIX;

  const int n0 = blockIdx.x * PIX_PER_WG;

  // Prologue: stage pixel n0 into buffer 0
  stage<2>(frame, x_lds[0], n0, tid);
  stage<8>(tmpl,  t_lds[0], n0, tid);
  stage_wait();
  __syncthreads();

  for (int p = 0; p < PIX_PER_WG; ++p) {
    const int n   = n0 + p;
    const int cur = p & 1;
    const int nxt = cur ^ 1;

    // Kick off next pixel's DMA immediately: overlaps sums + WMMA + reduce.
    if (p + 1 < PIX_PER_WG) {
      stage<2>(frame, x_lds[nxt], n + 1, tid);
      stage<8>(tmpl,  t_lds[nxt], n + 1, tid);
    }

    const float* xc = x_lds[cur];
    const float* tc = t_lds[cur];

    // ---- row sums of squares (80 rows, float4 reads, <=2-way bank conflict)
    if (tid < BDIM + KTPL) {
      const float4* r4 = (const float4*)((tid < BDIM)
                              ? (xc + tid * LDSW)
                              : (tc + (tid - BDIM) * LDSW));
      float s = 0.f;
#pragma unroll
      for (int q = 0; q < 16; ++q) {
        float4 v = r4[q];
        s += v.x * v.x + v.y * v.y + v.z * v.z + v.w * v.w;
      }
      sums[tid] = s;
    }
    __syncthreads();

    // ---- 16x16x64 fp32 WMMA: this wave's 16-template tile of X·Tᵀ
    // A (16x4 f32): VGPR0=K{0|2}, VGPR1=K{1|3} by lane half; B mirrors it.
    v8f acc = {};
    const float* ta = xc;
    const float* tb = tc + (wave * 16) * LDSW;
#pragma unroll
    for (int c = 0; c < 16; ++c) {
      const int d0 = c * 4 + hi * 2;
      v2f a, b;
      a[0] = ta[j * LDSW + d0];
      a[1] = ta[j * LDSW + d0 + 1];
      b[0] = tb[j * LDSW + d0];
      b[1] = tb[j * LDSW + d0 + 1];
      acc = __builtin_amdgcn_wmma_f32_16x16x4_f32(
              false, a, false, b, (short)0, acc, false, false);
    }

    // ---- scatter t2[k] - 2*xt into padded [b][k] matrix (conflict-free banks)
    // D layout: VGPR v, lane -> (b = v + 8*hi, k = wave*16 + j)
    const int   kidx = wave * 16 + j;
    const float t2v  = sums[BDIM + kidx];
#pragma unroll
    for (int v = 0; v < 8; ++v)
      dist_lds[(v + 8 * hi) * LDSW + kidx] = t2v - 2.0f * acc[v];
    __syncthreads();

    // ---- 16 lanes: contiguous b128 scan of 64 templates, min + first-index argmin
    if (tid < BDIM) {
      const float4* row4 = (const float4*)(dist_lds + tid * LDSW);
      float best = 3.4e38f;
      int   bidx = 0;
#pragma unroll
      for (int k4 = 0; k4 < 16; ++k4) {
        float4 dv = row4[k4];
        const int k = k4 * 4;
        if (dv.x < best) { best = dv.x; bidx = k;     }
        if (dv.y < best) { best = dv.y; bidx = k + 1; }
        if (dv.z < best) { best = dv.z; bidx = k + 2; }
        if (dv.w < best) { best = dv.w; bidx = k + 3; }
      }
      best += sums[tid];                       // + x2[b] (constant per row)
      const size_t o = (size_t)tid * NPIX + n;
      mind[o]  = best;
      pred[o]  = (float)tcls[bidx];
      mask0[o] = (best <= 0.5f) ? 1.0f : 0.0f;
      mask1[o] = (best <= 1.0f) ? 1.0f : 0.0f;
    }

    stage_wait();      // this wave's async DMA for buf[nxt] has landed in LDS
    __syncthreads();   // all waves: nxt fully staged, cur free to overwrite
  }
}

extern "C" void kernel_launch(void* const* d_in, const int* in_sizes, int n_in,
                              void* d_out, int out_size, void* d_ws, size_t ws_size,
                              hipStream_t stream) {
  (void)in_sizes; (void)n_in; (void)out_size; (void)d_ws; (void)ws_size;
  const float* frame = (const float*)d_in[0];
  const float* tmpl  = (const float*)d_in[1];
  const int*   tcls  = (const int*)d_in[2];
  float* out = (float*)d_out;

  dim3 grid(NPIX / PIX_PER_WG);   // 1024 workgroups
  dim3 block(THREADS);            // 128 threads = 4 wave32
  hipLaunchKernelGGL(openset_classify_kernel, grid, block, 0, stream,
                     frame, tmpl, tcls, out);
}